// GraphSAGE_62199716381241
// MI455X (gfx1250) — compile-verified
//
#include <hip/hip_runtime.h>

typedef __attribute__((ext_vector_type(2))) float v2f;
typedef __attribute__((ext_vector_type(8))) float v8f;

#define N_NODES  100000
#define N_EDGES  800000
#define IN_SIZE  128
#define HIDDEN   256
#define OUT_SIZE 128

// Workspace layout (in floats):
//   [0, 102400)                      deg (padded)
//   [NS_OFF, NS_OFF + N*256)         region1: nsum1 (first N*128) then nsum2 (N*256)
//   [H1_OFF, H1_OFF + N*256)         h1
#define DEG_OFF 0
#define NS_OFF  102400
#define H1_OFF  (NS_OFF + N_NODES * 256)

__device__ __forceinline__ void atomic_add_f32(float* p, float v) {
    // Relaxed agent-scope fp32 atomic -> global_atomic_add_f32 (no CAS loop)
    __hip_atomic_fetch_add(p, v, __ATOMIC_RELAXED, __HIP_MEMORY_SCOPE_AGENT);
}

__global__ void zero_kernel(float4* __restrict__ p, long long n4) {
    long long i = blockIdx.x * (long long)blockDim.x + threadIdx.x;
    long long stride = (long long)gridDim.x * blockDim.x;
    float4 z = {0.f, 0.f, 0.f, 0.f};
    for (; i < n4; i += stride) p[i] = z;
}

__global__ void degree_kernel(const int* __restrict__ dst, float* __restrict__ deg) {
    int i = blockIdx.x * blockDim.x + threadIdx.x;
    int stride = gridDim.x * blockDim.x;
    for (; i < N_EDGES; i += stride)
        atomic_add_f32(&deg[dst[i]], 1.0f);
}

// Edge-parallel scatter: each thread owns one (edge, 4-channel group).
// float4 coalesced gather from feat[src], 4 fp32 atomics into nsum[dst].
template <int D>
__global__ void scatter_kernel(const float* __restrict__ feat, const int* __restrict__ src,
                               const int* __restrict__ dst, float* __restrict__ nsum) {
    constexpr int GPE = D / 4;  // float4 groups per edge (power of two)
    const long long total = (long long)N_EDGES * GPE;
    long long i = blockIdx.x * (long long)blockDim.x + threadIdx.x;
    long long stride = (long long)gridDim.x * blockDim.x;
    for (; i < total; i += stride) {
        int e = (int)(i / GPE);
        int g = (int)(i % GPE);
        int s = src[e];
        int d = dst[e];
        const float4 v = *(const float4*)(feat + (size_t)s * D + g * 4);
        float* o = nsum + (size_t)d * D + g * 4;
        atomic_add_f32(o + 0, v.x);
        atomic_add_f32(o + 1, v.y);
        atomic_add_f32(o + 2, v.z);
        atomic_add_f32(o + 3, v.w);
    }
}

// Fused: A = (nsum + hin) * 1/(deg+1), then D = A @ W + bias (optional ReLU)
// using V_WMMA_F32_16X16X4_F32. Block = 128 threads = 4 waves, one 16-row M-tile.
// Each wave owns TPW 16-col output tiles (wave N-slab = TPW*16 columns).
template <int K, int NOUT, int TPW, bool RELU>
__global__ __launch_bounds__(128) void gemm_agg_kernel(
    const float* __restrict__ hin, const float* __restrict__ nsum,
    const float* __restrict__ deg, const float* __restrict__ W,
    const float* __restrict__ bias, float* __restrict__ hout) {
    // +4 float padding: 16 strided rows map to disjoint LDS bank pairs for ds_load_b64
    __shared__ float As[16][K + 4];
    __shared__ float rdeg_s[16];

    const int tid  = threadIdx.x;
    const int row0 = blockIdx.x * 16;

    if (tid < 16) rdeg_s[tid] = 1.0f / (deg[row0 + tid] + 1.0f);
    __syncthreads();

    // Stage normalized A-tile (16 x K) into LDS; coalesced global reads.
    for (int i = tid; i < 16 * K; i += 128) {
        int m = i / K;
        int k = i - m * K;
        size_t idx = (size_t)(row0 + m) * K + k;
        As[m][k] = (nsum[idx] + hin[idx]) * rdeg_s[m];
    }
    __syncthreads();

    const int lane = tid & 31;
    const int wave = tid >> 5;
    const int m    = lane & 15;   // row (A) / col (B,D) within tile
    const int khi  = lane >> 4;   // K-pair select: lanes 16-31 hold K+2,K+3
    const int nb   = wave * (TPW * 16);

    v8f acc[TPW] = {};

    for (int k0 = 0; k0 < K; k0 += 4) {
        // A fragment: lane (m + 16*khi) holds A[m][k0+2*khi + {0,1}]
        v2f a = *(const v2f*)&As[m][k0 + 2 * khi];
#pragma unroll
        for (int t = 0; t < TPW; ++t) {
            // B fragment: lane (n + 16*khi) holds B[k0+2*khi + {0,1}][n], n = m
            const float* wp = W + (size_t)(k0 + 2 * khi) * NOUT + (nb + 16 * t + m);
            v2f b;
            b.x = wp[0];
            b.y = wp[NOUT];
            acc[t] = __builtin_amdgcn_wmma_f32_16x16x4_f32(
                false, a, false, b, (short)0, acc[t], false, false);
        }
    }

    // D layout: VGPR r -> row r (lanes 0-15) / row r+8 (lanes 16-31), col = n
    const int rbase = khi ? 8 : 0;
#pragma unroll
    for (int t = 0; t < TPW; ++t) {
        int col  = nb + 16 * t + m;
        float bv = bias[col];
#pragma unroll
        for (int r = 0; r < 8; ++r) {
            float v = acc[t][r] + bv;
            if (RELU) v = v > 0.0f ? v : 0.0f;
            hout[(size_t)(row0 + rbase + r) * NOUT + col] = v;
        }
    }
}

extern "C" void kernel_launch(void* const* d_in, const int* in_sizes, int n_in,
                              void* d_out, int out_size, void* d_ws, size_t ws_size,
                              hipStream_t stream) {
    const float* feat = (const float*)d_in[0];
    const int*   src  = (const int*)d_in[1];
    const int*   dst  = (const int*)d_in[2];
    const float* W1   = (const float*)d_in[3];
    const float* b1   = (const float*)d_in[4];
    const float* W2   = (const float*)d_in[5];
    const float* b2   = (const float*)d_in[6];

    float* ws      = (float*)d_ws;
    float* deg     = ws + DEG_OFF;
    float* region1 = ws + NS_OFF;  // nsum1, later nsum2
    float* h1      = ws + H1_OFF;
    float* out     = (float*)d_out;

    // Phase 0: zero deg + nsum1
    zero_kernel<<<2048, 256, 0, stream>>>((float4*)ws,
        (long long)(NS_OFF + N_NODES * IN_SIZE) / 4);
    // Phase 1: in-degree
    degree_kernel<<<1024, 256, 0, stream>>>(dst, deg);
    // Phase 2: scatter layer-1 neighbor sums
    scatter_kernel<IN_SIZE><<<4096, 256, 0, stream>>>(feat, src, dst, region1);
    // Phase 3: h1 = relu(((nsum1 + feat)/(deg+1)) @ W1 + b1)   [WMMA]
    gemm_agg_kernel<IN_SIZE, HIDDEN, 4, true>
        <<<N_NODES / 16, 128, 0, stream>>>(feat, region1, deg, W1, b1, h1);
    // Phase 4: re-zero region1 as nsum2
    zero_kernel<<<2048, 256, 0, stream>>>((float4*)region1,
        (long long)N_NODES * HIDDEN / 4);
    // Phase 5: scatter layer-2 neighbor sums
    scatter_kernel<HIDDEN><<<8192, 256, 0, stream>>>(h1, src, dst, region1);
    // Phase 6: out = ((nsum2 + h1)/(deg+1)) @ W2 + b2          [WMMA]
    gemm_agg_kernel<HIDDEN, OUT_SIZE, 2, false>
        <<<N_NODES / 16, 128, 0, stream>>>(h1, region1, deg, W2, b2, out);
}